// NeighborlistForInference_22960895164668
// MI455X (gfx1250) — compile-verified
//
#include <hip/hip_runtime.h>
#include <stdint.h>

// Pairwise neighbor computation for brute_nsq unique-pair list with orthorhombic PBC.
// Outputs concatenated into d_out (float32):
//   [0,2P)   pair_indices (i row then j row, as float values)
//   [2P,3P)  d_ij masked
//   [3P,6P)  r_ij masked, row-major [P,3]
//   [6P,7P)  in_cutoff mask as 0.0/1.0
//
// Store-bandwidth bound (~680 MB/call => ~29 us at 23.3 TB/s). Positions are
// staged into LDS once per block via CDNA5 async global->LDS DMA; all big
// outputs use 128-bit non-temporal stores.

typedef float v4f __attribute__((ext_vector_type(4)));
typedef int   v4i __attribute__((ext_vector_type(4)));

#define TPB 256
#define CUTOFF 0.5f

__device__ __forceinline__ float wrap_pbc(float r, float L, float invL, float hL) {
    // jnp.remainder(r + L/2, L) - L/2
    float t = r + hL;
    t -= L * floorf(t * invL);
    return t - hL;
}

extern "C" __global__ __launch_bounds__(TPB)
void nlist_pairs_kernel(const float* __restrict__ pos,
                        const float* __restrict__ box,
                        const int* __restrict__ ipairs,
                        const int* __restrict__ jpairs,
                        float* __restrict__ out,
                        int N, long long P)
{
    extern __shared__ float sP[];  // 4*N floats: particle k at sP[4k..4k+2], w unused

    // ---- Stage positions into LDS via CDNA5 async global->LDS DMA ----
    // b128 reads 16B starting at the 12B-strided xyz triple (4th dword is the
    // next particle's x; harmless). Last particle done with b64+b32 to stay
    // inside the buffer.
    for (int k = threadIdx.x; k < N - 1; k += TPB) {
        unsigned lds = (unsigned)(uintptr_t)(&sP[4 * k]);
        unsigned long long g = (unsigned long long)(uintptr_t)(pos + 3 * k);
        asm volatile("global_load_async_to_lds_b128 %0, %1, off"
                     :: "v"(lds), "v"(g) : "memory");
    }
    if (threadIdx.x == 0 && N > 0) {
        int k = N - 1;
        unsigned lds = (unsigned)(uintptr_t)(&sP[4 * k]);
        unsigned long long g = (unsigned long long)(uintptr_t)(pos + 3 * k);
        asm volatile("global_load_async_to_lds_b64 %0, %1, off"
                     :: "v"(lds), "v"(g) : "memory");
        asm volatile("global_load_async_to_lds_b32 %0, %1, off"
                     :: "v"(lds + 8u), "v"(g + 8ull) : "memory");
    }
    asm volatile("s_wait_asynccnt 0" ::: "memory");
    __syncthreads();

    const float Lx = box[0], Ly = box[4], Lz = box[8];
    const float iLx = 1.0f / Lx, iLy = 1.0f / Ly, iLz = 1.0f / Lz;
    const float hLx = 0.5f * Lx, hLy = 0.5f * Ly, hLz = 0.5f * Lz;

    float* __restrict__ o_i = out;
    float* __restrict__ o_j = out + P;
    float* __restrict__ o_d = out + 2 * P;
    float* __restrict__ o_r = out + 3 * P;
    float* __restrict__ o_m = out + 6 * P;

    const long long nvec   = (P + 3) >> 2;                 // 4 pairs per thread-iter
    const long long stride = (long long)gridDim.x * TPB;

    for (long long v = (long long)blockIdx.x * TPB + threadIdx.x; v < nvec; v += stride) {
        const long long p = v << 2;
        if (p + 4 <= P) {
            // Speculative prefetch of the streaming index arrays well ahead
            // (OOB prefetches are silently dropped on gfx1250).
            __builtin_prefetch(ipairs + p + stride * 16, 0, 0);
            __builtin_prefetch(jpairs + p + stride * 16, 0, 0);

            const v4i i4 = __builtin_nontemporal_load((const v4i*)(ipairs + p));
            const v4i j4 = __builtin_nontemporal_load((const v4i*)(jpairs + p));

            v4f fi, fj, fd, fm;
            float rr[12];
#pragma unroll
            for (int e = 0; e < 4; ++e) {
                const int i = i4[e];
                const int j = j4[e];
                const v4f Pi = *(const v4f*)(sP + 4 * i);  // ds_load_b128
                const v4f Pj = *(const v4f*)(sP + 4 * j);
                float dx = wrap_pbc(Pi.x - Pj.x, Lx, iLx, hLx);
                float dy = wrap_pbc(Pi.y - Pj.y, Ly, iLy, hLy);
                float dz = wrap_pbc(Pi.z - Pj.z, Lz, iLz, hLz);
                const float d  = sqrtf(dx * dx + dy * dy + dz * dz);
                const bool in  = (d <= CUTOFF);
                fi[e] = (float)i;
                fj[e] = (float)j;
                fd[e] = in ? d : 0.0f;
                fm[e] = in ? 1.0f : 0.0f;
                rr[3 * e + 0] = in ? dx : 0.0f;
                rr[3 * e + 1] = in ? dy : 0.0f;
                rr[3 * e + 2] = in ? dz : 0.0f;
            }
            __builtin_nontemporal_store(fi, (v4f*)(o_i + p));
            __builtin_nontemporal_store(fj, (v4f*)(o_j + p));
            __builtin_nontemporal_store(fd, (v4f*)(o_d + p));
            const v4f r0 = {rr[0], rr[1], rr[2],  rr[3]};
            const v4f r1 = {rr[4], rr[5], rr[6],  rr[7]};
            const v4f r2 = {rr[8], rr[9], rr[10], rr[11]};
            __builtin_nontemporal_store(r0, (v4f*)(o_r + 3 * p));
            __builtin_nontemporal_store(r1, (v4f*)(o_r + 3 * p + 4));
            __builtin_nontemporal_store(r2, (v4f*)(o_r + 3 * p + 8));
            __builtin_nontemporal_store(fm, (v4f*)(o_m + p));
        } else {
            // scalar tail (only if P % 4 != 0)
            for (int e = 0; e < 4; ++e) {
                const long long q = p + e;
                if (q >= P) break;
                const int i = ipairs[q];
                const int j = jpairs[q];
                const v4f Pi = *(const v4f*)(sP + 4 * i);
                const v4f Pj = *(const v4f*)(sP + 4 * j);
                float dx = wrap_pbc(Pi.x - Pj.x, Lx, iLx, hLx);
                float dy = wrap_pbc(Pi.y - Pj.y, Ly, iLy, hLy);
                float dz = wrap_pbc(Pi.z - Pj.z, Lz, iLz, hLz);
                const float d = sqrtf(dx * dx + dy * dy + dz * dz);
                const bool in = (d <= CUTOFF);
                o_i[q] = (float)i;
                o_j[q] = (float)j;
                o_d[q] = in ? d : 0.0f;
                o_r[3 * q + 0] = in ? dx : 0.0f;
                o_r[3 * q + 1] = in ? dy : 0.0f;
                o_r[3 * q + 2] = in ? dz : 0.0f;
                o_m[q] = in ? 1.0f : 0.0f;
            }
        }
    }
}

extern "C" void kernel_launch(void* const* d_in, const int* in_sizes, int n_in,
                              void* d_out, int out_size, void* d_ws, size_t ws_size,
                              hipStream_t stream) {
    (void)n_in; (void)out_size; (void)d_ws; (void)ws_size;
    const float* pos = (const float*)d_in[0];   // [N,3] float32
    const float* box = (const float*)d_in[1];   // [3,3] float32
    const int*   ip  = (const int*)d_in[2];     // [P] int32
    const int*   jp  = (const int*)d_in[3];     // [P] int32
    float* out = (float*)d_out;

    const int       N = in_sizes[0] / 3;
    const long long P = (long long)in_sizes[2];

    const long long nvec = (P + 3) / 4;
    long long need_blocks = (nvec + TPB - 1) / TPB;
    int blocks = 1024;                          // grid-stride; amortizes LDS staging
    if (need_blocks < blocks) blocks = (int)need_blocks;
    if (blocks < 1) blocks = 1;

    const size_t shmem = (size_t)N * 4 * sizeof(float);  // 96 KB for N=6144

    hipLaunchKernelGGL(nlist_pairs_kernel, dim3(blocks), dim3(TPB), shmem, stream,
                       pos, box, ip, jp, out, N, P);
}